// ShadingLoss_44435731645075
// MI455X (gfx1250) — compile-verified
//
#include <hip/hip_runtime.h>
#include <hip/hip_bf16.h>

#define H_IMG 2160
#define W_IMG 3840
#define TILE  32
#define REG   34              // render/rgb/mask logical region = TILE + 2
#define DREG  35              // depth region = TILE + 3
#define RSTR  36              // LDS row stride (floats); cols 34,35 are dead pad
#define NSLOT (REG * RSTR)    // 1224 region slots (incl. pad cols)
#define NGRP  ((NSLOT + 15) / 16)   // 77 WMMA groups of 16 slots
#define NWR   (NGRP * 16)     // 1232 writable render slots

typedef __attribute__((ext_vector_type(2))) float v2f;
typedef __attribute__((ext_vector_type(8))) float v8f;

// D = A(16x4) * B(4x16) + C(16x16), all f32, wave32.
__device__ __forceinline__ v8f wmma4(v2f a, v2f b, v8f c) {
    return __builtin_amdgcn_wmma_f32_16x16x4_f32(
        /*neg_a=*/false, a, /*neg_b=*/false, b,
        /*c_mod=*/(short)0, c, /*reuse_a=*/false, /*reuse_b=*/false);
}

__global__ void sl_init(double* wsum, float* wcnt) {
    *wsum = 0.0;
    *wcnt = 0.0f;
}

__global__ void __launch_bounds__(256)
sl_main(const float* __restrict__ depth,
        const float* __restrict__ rgb,
        const int*   __restrict__ mask,
        const float* __restrict__ lmat,   // [9][3]
        const float* __restrict__ K,      // [3][3]
        double* __restrict__ wsum,
        float*  __restrict__ wcnt) {
    __shared__ float sdepth[DREG * RSTR];              // 35 rows x 36 cols
    __shared__ float smaskf[NSLOT];
    __shared__ __align__(16) float srender[3][NWR];
    __shared__ __align__(16) float srgb[3][NWR];
    __shared__ float rsum[256];
    __shared__ float rcnt[256];

    const int tid = threadIdx.x;
    const int tx0 = blockIdx.x * TILE;
    const int ty0 = blockIdx.y * TILE;

    const float fx = K[0], cx = K[2], fy = K[4], cy = K[5];
    const float ifx = 1.0f / fx, ify = 1.0f / fy;

    // ---- Stage tile + halo into LDS (zero outside image = SAME zero-pad) ----
    for (int i = tid; i < DREG * RSTR; i += 256) {      // depth: 35 rows x 36
        int rr = i / RSTR, cc = i % RSTR;
        int gy = ty0 - 1 + rr, gx = tx0 - 1 + cc;
        float d = 0.f;
        if (cc < DREG && gy >= 0 && gy < H_IMG && gx >= 0 && gx < W_IMG)
            d = depth[(size_t)gy * W_IMG + gx];
        sdepth[i] = d;
    }
    for (int i = tid; i < NSLOT; i += 256) {            // mask+rgb: 34 rows x 36
        int rr = i / RSTR, cc = i % RSTR;
        int gy = ty0 - 1 + rr, gx = tx0 - 1 + cc;
        float m = 0.f, r0 = 0.f, r1 = 0.f, r2 = 0.f;
        if (cc < REG && gy >= 0 && gy < H_IMG && gx >= 0 && gx < W_IMG) {
            m = (mask[(size_t)gy * W_IMG + gx] > 0) ? 1.f : 0.f;
            const float* rp = rgb + ((size_t)gy * W_IMG + gx) * 3;
            r0 = rp[0]; r1 = rp[1]; r2 = rp[2];
        }
        smaskf[i] = m;
        srgb[0][i] = r0; srgb[1][i] = r1; srgb[2][i] = r2;
    }
    __syncthreads();

    // ---- Phase A: normals + SH basis; render = mask * (Hsh . l) via WMMA ----
    const int wave = tid >> 5;
    const int lane = tid & 31;
    const int half = lane >> 4;   // K-half owned by this lane (f32 A layout)
    const int m16  = lane & 15;

    // Loop-invariant B fragments: B(k,n) = l[k][n] for n<3, K padded 9->12.
    // f32 B 4x16 layout: lane = n + 16*(k>=2), vgpr = k%2.
    v2f bf[3];
    #pragma unroll
    for (int c2 = 0; c2 < 3; ++c2) {
        float b0 = 0.f, b1 = 0.f;
        int k0 = c2 * 4 + half * 2;
        if (m16 < 3) {
            if (k0     < 9) b0 = lmat[k0 * 3 + m16];
            if (k0 + 1 < 9) b1 = lmat[(k0 + 1) * 3 + m16];
        }
        v2f t; t.x = b0; t.y = b1; bf[c2] = t;
    }

    for (int g = wave; g < NGRP; g += 8) {
        int p  = g * 16 + m16;             // both lane-halves: same 16 slots
        int sc = (p < NSLOT) ? p : (NSLOT - 1);   // clamp => rr<=33, all LDS in-bounds
        int rr = sc / RSTR, cc = sc - rr * RSTR;
        int gy = ty0 - 1 + rr, gx = tx0 - 1 + cc;

        float d0 = sdepth[sc];
        float d1 = sdepth[sc + 1];
        float d2 = sdepth[sc + RSTR];
        float rx0 = ((float)gx - cx) * ifx;
        float ry0 = ((float)gy - cy) * ify;
        float rx1 = ((float)(gx + 1) - cx) * ifx;
        float ry1 = ((float)(gy + 1) - cy) * ify;

        float px = d0 * rx0, py = d0 * ry0, pz = d0;
        float plx = 0.f, ply = 0.f, plz = 0.f;   // zero-padded neighbors
        float pux = 0.f, puy = 0.f, puz = 0.f;
        if (gx + 1 < W_IMG) { plx = d1 * rx1; ply = d1 * ry0; plz = d1; }
        if (gy + 1 < H_IMG) { pux = d2 * rx0; puy = d2 * ry1; puz = d2; }

        float ax = plx - px, ay = ply - py, az = plz - pz;
        float bx = pux - px, by = puy - py, bz = puz - pz;
        float nx = ay * bz - az * by;
        float ny = az * bx - ax * bz;
        float nz = ax * by - ay * bx;
        float dot = nx * nx + ny * ny + nz * nz;
        float iv  = __builtin_amdgcn_rsqf(fmaxf(dot, 1e-24f));  // v_rsq_f32
        nx *= iv; ny *= iv; nz *= iv;

        float mv = smaskf[sc];       // fold mask into Hsh -> D is masked render
        float hs0 = mv;
        float hs1 = mv * ny,  hs2 = mv * nz,  hs3 = mv * nx;
        float hs4 = mv * nx * ny, hs5 = mv * ny * nz;
        float hs6 = mv * (-nx * nx - ny * ny + 2.f * nz * nz);
        float hs7 = mv * nz * nx, hs8 = mv * (nx * nx - ny * ny);

        // A fragments: A(m,k) -> lane (k>=2 ? 16+m : m), vgpr k%2
        v2f a0, a1, a2;
        a0.x = half ? hs2 : hs0;  a0.y = half ? hs3 : hs1;
        a1.x = half ? hs6 : hs4;  a1.y = half ? hs7 : hs5;
        a2.x = half ? 0.f : hs8;  a2.y = 0.f;    // K = 9..11 zero pad

        v8f acc = {0.f, 0.f, 0.f, 0.f, 0.f, 0.f, 0.f, 0.f};
        acc = wmma4(a0, bf[0], acc);
        acc = wmma4(a1, bf[1], acc);
        acc = wmma4(a2, bf[2], acc);

        // D(m,n): lane n (= channel) holds rows m = half*8 + v in acc[v].
        // Slot-linear layout => 8 consecutive dwords, unconditional (b128-merge).
        if (m16 < 3) {
            float* dst = &srender[m16][g * 16 + half * 8];
            #pragma unroll
            for (int v = 0; v < 8; ++v) dst[v] = acc[v];
        }
    }
    __syncthreads();

    // ---- Phase B: 3x3 box means, masked squared diff, block reduce ----
    float ls = 0.f, lc = 0.f;
    for (int i = tid; i < TILE * TILE; i += 256) {
        int oy = i >> 5, ox = i & 31;
        int base = (oy + 1) * RSTR + (ox + 1);
        float m = smaskf[base];
        if (m > 0.f) {
            float dsum = 0.f;
            #pragma unroll
            for (int ch = 0; ch < 3; ++ch) {
                const float* sr = srender[ch];
                const float* sg = srgb[ch];
                float rm = 0.f, gm = 0.f;
                #pragma unroll
                for (int dy = -1; dy <= 1; ++dy)
                    #pragma unroll
                    for (int dx = -1; dx <= 1; ++dx) {
                        int o = base + dy * RSTR + dx;
                        rm += sr[o];
                        gm += sg[o];
                    }
                rm *= (1.f / 9.f);
                gm *= (1.f / 9.f);
                float d = (sr[base] - rm) - (sg[base] - gm);
                dsum += d * d;
            }
            ls += dsum;
            lc += 1.f;
        }
    }
    rsum[tid] = ls;
    rcnt[tid] = lc;
    __syncthreads();
    for (int s = 128; s > 0; s >>= 1) {
        if (tid < s) {
            rsum[tid] += rsum[tid + s];
            rcnt[tid] += rcnt[tid + s];
        }
        __syncthreads();
    }
    if (tid == 0) {
        atomicAdd(wsum, (double)rsum[0]);
        atomicAdd(wcnt, rcnt[0]);
    }
}

__global__ void sl_fin(const double* wsum, const float* wcnt, float* out) {
    double c = (double)(*wcnt) * 3.0;
    out[0] = (c > 0.0) ? (float)(*wsum / c) : 0.0f;
}

extern "C" void kernel_launch(void* const* d_in, const int* in_sizes, int n_in,
                              void* d_out, int out_size, void* d_ws, size_t ws_size,
                              hipStream_t stream) {
    (void)in_sizes; (void)n_in; (void)out_size; (void)ws_size;
    const float* depth = (const float*)d_in[0];
    const float* rgb   = (const float*)d_in[1];
    const int*   mask  = (const int*)d_in[2];
    const float* lmat  = (const float*)d_in[3];
    const float* K     = (const float*)d_in[4];

    double* wsum = (double*)d_ws;
    float*  wcnt = (float*)((char*)d_ws + sizeof(double));

    sl_init<<<1, 1, 0, stream>>>(wsum, wcnt);
    dim3 grid((W_IMG + TILE - 1) / TILE, (H_IMG + TILE - 1) / TILE);
    sl_main<<<grid, 256, 0, stream>>>(depth, rgb, mask, lmat, K, wsum, wcnt);
    sl_fin<<<1, 1, 0, stream>>>(wsum, wcnt, (float*)d_out);
}